// GAT_23931557773432
// MI455X (gfx1250) — compile-verified
//
#include <hip/hip_runtime.h>

#define NEG_SLOPE 0.2f

typedef __attribute__((ext_vector_type(2))) float v2f;
typedef __attribute__((ext_vector_type(8))) float v8f;

// ---- monotonic float <-> uint key for atomic segment-max -------------------
__device__ __forceinline__ unsigned fkey(float f) {
    unsigned b = __float_as_uint(f);
    return (b & 0x80000000u) ? ~b : (b | 0x80000000u);
}
__device__ __forceinline__ float fdec(unsigned k) {
    unsigned b = (k & 0x80000000u) ? (k & 0x7FFFFFFFu) : ~k;
    return __uint_as_float(b);
}

// ---- h[n,M] = x[n,K] @ w[K,M], one wave -> one 16x16 tile, fp32 WMMA -------
// A layout (16x4 f32): lane<16 row=lane holds K={k0,k0+1}; lane>=16 K={k0+2,k0+3}
// B layout (4x16 f32): symmetric, lanes hold N, VGPRs hold K
// C layout (16x16 f32): VGPR r, lane L -> row r+8*(L>=16), col L%16
__global__ void gemm_wmma_f32(const float* __restrict__ x,
                              const float* __restrict__ w,
                              float* __restrict__ h, int K, int M) {
    const int row0 = blockIdx.x * 16;
    const int col0 = (blockIdx.y * blockDim.y + threadIdx.y) * 16;
    const int lane = threadIdx.x;
    const int half = lane >> 4;   // 0 or 1
    const int lr   = lane & 15;

    const float* xrow = x + (size_t)(row0 + lr) * K;
    v8f c = {};
    for (int k0 = 0; k0 < K; k0 += 4) {
        const int k = k0 + 2 * half;
        v2f a = *(const v2f*)(xrow + k);              // 8B aligned (k even)
        v2f b;
        b.x = w[(size_t)k * M + col0 + lr];
        b.y = w[(size_t)(k + 1) * M + col0 + lr];
        c = __builtin_amdgcn_wmma_f32_16x16x4_f32(
                /*neg_a=*/false, a, /*neg_b=*/false, b,
                /*c_mod=*/(short)0, c, /*reuse_a=*/false, /*reuse_b=*/false);
    }
#pragma unroll
    for (int r = 0; r < 8; r++) {
        h[(size_t)(row0 + r + 8 * half) * M + col0 + lr] = c[r];
    }
}

// ---- es[n,h] = sum_f h[n,h,f]*a_src[h,f] ; ed likewise ---------------------
__global__ void node_scores(const float* __restrict__ h,
                            const float* __restrict__ asrc,
                            const float* __restrict__ adst,
                            float* __restrict__ es, float* __restrict__ ed,
                            int NH, int H, int F) {
    int i = blockIdx.x * blockDim.x + threadIdx.x;
    if (i >= NH) return;
    int head = i % H;
    const float* hp = h + (size_t)i * F;   // i == n*H+head, h is [n,H,F]
    const float* a1 = asrc + head * F;
    const float* a2 = adst + head * F;
    float s1 = 0.f, s2 = 0.f;
#pragma unroll
    for (int f = 0; f < 16; f++) { float v = hp[f]; s1 += v * a1[f]; s2 += v * a2[f]; }
    es[i] = s1; ed[i] = s2;
}

// ---- pass 1: segment max over dst neighborhoods ----------------------------
__global__ void edge_max(const int* __restrict__ A0, const int* __restrict__ A1,
                         int E, int Etot,
                         const float* __restrict__ es, const float* __restrict__ ed,
                         unsigned* __restrict__ m, int H) {
    int e = blockIdx.x * blockDim.x + threadIdx.x;
    if (e >= Etot) return;
    int s = (e < E) ? A0[e] : (e - E);   // self-loops appended
    int d = (e < E) ? A1[e] : (e - E);
    for (int hd = 0; hd < H; hd++) {
        float v = es[s * H + hd] + ed[d * H + hd];
        v = (v > 0.f) ? v : NEG_SLOPE * v;
        atomicMax(&m[d * H + hd], fkey(v));
    }
}

// ---- pass 2: exp, segment sum, weighted scatter-add ------------------------
__global__ void edge_acc(const int* __restrict__ A0, const int* __restrict__ A1,
                         int E, int Etot,
                         const float* __restrict__ es, const float* __restrict__ ed,
                         const unsigned* __restrict__ m, float* __restrict__ ssum,
                         const float* __restrict__ h, float* __restrict__ acc,
                         int H, int F) {
    int i = blockIdx.x * blockDim.x + threadIdx.x;
    int tot = Etot * H;
    if (i >= tot) return;
    int e  = i / H;
    int hd = i % H;
    int s = (e < E) ? A0[e] : (e - E);
    int d = (e < E) ? A1[e] : (e - E);
    float v = es[s * H + hd] + ed[d * H + hd];
    v = (v > 0.f) ? v : NEG_SLOPE * v;
    float ex = __expf(v - fdec(m[d * H + hd]));
    atomicAdd(&ssum[d * H + hd], ex);
    const float* hs = h   + ((size_t)s * H + hd) * F;
    float*       ap = acc + ((size_t)d * H + hd) * F;
#pragma unroll
    for (int f = 0; f < 16; f++) atomicAdd(&ap[f], hs[f] * ex);
}

// ---- finalize: concat heads, +bias, ReLU -----------------------------------
__global__ void finalize_concat(const float* __restrict__ acc,
                                const float* __restrict__ ssum,
                                const float* __restrict__ bias,
                                float* __restrict__ xout, int Nn, int H, int F) {
    int i = blockIdx.x * blockDim.x + threadIdx.x;
    int HF = H * F;
    if (i >= Nn * HF) return;
    int n = i / HF, col = i % HF, hd = col / F;
    float v = acc[i] / (ssum[n * H + hd] + 1e-16f) + bias[col];
    xout[i] = v > 0.f ? v : 0.f;
}

// ---- finalize layer 3: mean over heads, +bias ------------------------------
__global__ void finalize_mean(const float* __restrict__ acc,
                              const float* __restrict__ ssum,
                              const float* __restrict__ bias,
                              float* __restrict__ out, int Nn, int H, int F) {
    int i = blockIdx.x * blockDim.x + threadIdx.x;
    if (i >= Nn * F) return;
    int n = i / F, f = i % F;
    float sum = 0.f;
    for (int hd = 0; hd < H; hd++)
        sum += acc[((size_t)n * H + hd) * F + f] / (ssum[n * H + hd] + 1e-16f);
    out[i] = sum / (float)H + bias[f];
}

extern "C" void kernel_launch(void* const* d_in, const int* in_sizes, int n_in,
                              void* d_out, int out_size, void* d_ws, size_t ws_size,
                              hipStream_t stream) {
    const int DIN = 128;
    const int Nn  = in_sizes[0] / DIN;      // 50000
    const int E   = in_sizes[1] / 2;        // 1600000
    const int Etot = E + Nn;

    const float* X  = (const float*)d_in[0];
    const int*   A0 = (const int*)d_in[1];
    const int*   A1 = A0 + E;

    const float* W[3]  = { (const float*)d_in[2], (const float*)d_in[6],  (const float*)d_in[10] };
    const float* AS[3] = { (const float*)d_in[3], (const float*)d_in[7],  (const float*)d_in[11] };
    const float* AD[3] = { (const float*)d_in[4], (const float*)d_in[8],  (const float*)d_in[12] };
    const float* B[3]  = { (const float*)d_in[5], (const float*)d_in[9],  (const float*)d_in[13] };
    const int Hh[3] = { 8, 8, 4 };
    const int Ff    = 16;
    const int Kin[3] = { 128, 128, 128 };
    const int Mout[3] = { 128, 128, 64 };

    // workspace layout
    float*    h    = (float*)d_ws;                     // Nn*128
    float*    xbuf = h    + (size_t)Nn * 128;          // Nn*128
    float*    acc  = xbuf + (size_t)Nn * 128;          // Nn*128
    float*    es   = acc  + (size_t)Nn * 128;          // Nn*8
    float*    ed   = es   + (size_t)Nn * 8;            // Nn*8
    unsigned* mkey = (unsigned*)(ed + (size_t)Nn * 8); // Nn*8
    float*    ssum = (float*)(mkey + (size_t)Nn * 8);  // Nn*8

    const float* xin = X;
    for (int L = 0; L < 3; L++) {
        const int H = Hh[L], F = Ff, K = Kin[L], M = Mout[L];
        const int HF = H * F;

        // zero accumulators (graph-capture safe)
        hipMemsetAsync(acc,  0, (size_t)Nn * HF * sizeof(float), stream);
        hipMemsetAsync(mkey, 0, (size_t)Nn * H  * sizeof(unsigned), stream);
        hipMemsetAsync(ssum, 0, (size_t)Nn * H  * sizeof(float), stream);

        // 1) h = xin @ W  via fp32 WMMA (Nn % 16 == 0, M % 64 == 0 or == 64)
        {
            dim3 blk(32, 4);
            dim3 grd(Nn / 16, M / 64);
            gemm_wmma_f32<<<grd, blk, 0, stream>>>(xin, W[L], h, K, M);
        }
        // 2) per-(node,head) attention scores
        {
            int NH = Nn * H;
            node_scores<<<(NH + 255) / 256, 256, 0, stream>>>(h, AS[L], AD[L], es, ed, NH, H, F);
        }
        // 3) segment max
        edge_max<<<(Etot + 255) / 256, 256, 0, stream>>>(A0, A1, E, Etot, es, ed, mkey, H);
        // 4) exp + segment sum + weighted aggregation
        {
            int tot = Etot * H;
            edge_acc<<<(tot + 255) / 256, 256, 0, stream>>>(A0, A1, E, Etot, es, ed, mkey,
                                                            ssum, h, acc, H, F);
        }
        // 5) finalize
        if (L < 2) {
            int tot = Nn * HF;
            finalize_concat<<<(tot + 255) / 256, 256, 0, stream>>>(acc, ssum, B[L], xbuf, Nn, H, F);
            xin = xbuf;
        } else {
            int tot = Nn * F;
            finalize_mean<<<(tot + 255) / 256, 256, 0, stream>>>(acc, ssum, B[L], (float*)d_out, Nn, H, F);
        }
    }
}